// refine_Module_22393959482069
// MI455X (gfx1250) — compile-verified
//
#include <hip/hip_runtime.h>
#include <hip/hip_bf16.h>

typedef float v2f __attribute__((ext_vector_type(2)));
typedef float v8f __attribute__((ext_vector_type(8)));

#define NN 3200              // NODE * FRAME = 25 * 128
#define NBATCH 16
#define TILES_PER_BATCH (NN / 16)   // 200
#define WAVES_PER_BLOCK 8
#define BLOCK_THREADS 256

// ---------------------------------------------------------------------------
// Kernel 1: per-batch min/max of x (one wave per batch).
// Rank-1 structure lets us derive every softmax row-max analytically from
// these two scalars, so we never need an O(N^2) max pass.
// ---------------------------------------------------------------------------
__global__ __launch_bounds__(32) void rowstat_kernel(const float* __restrict__ x,
                                                     float* __restrict__ bounds) {
    const int b = blockIdx.x;
    const int lane = threadIdx.x;
    const float* xb = x + b * NN;
    float mn = 3.4e38f, mx = -3.4e38f;
    for (int i = lane; i < NN; i += 32) {
        float v = xb[i];
        mn = fminf(mn, v);
        mx = fmaxf(mx, v);
    }
#pragma unroll
    for (int o = 16; o > 0; o >>= 1) {
        mn = fminf(mn, __shfl_xor(mn, o, 32));
        mx = fmaxf(mx, __shfl_xor(mx, o, 32));
    }
    if (lane == 0) {
        bounds[2 * b + 0] = mn;
        bounds[2 * b + 1] = mx;
    }
}

// ---------------------------------------------------------------------------
// Kernel 2: rank-1 softmax attention.
// Each wave owns a 16-row m-tile. Inner loop over n in chunks of 4:
//   A(16x4)   = exp(t_m * c_n - rowmax_m)           (2 exps per lane)
//   B(4x16)   = col0: x_n, col1: 1.0, rest 0
//   D += A*B  via v_wmma_f32_16x16x4_f32  -> D[:,0]=sum(p*x), D[:,1]=sum(p)
// Full f32 throughout; exp (v_exp_f32) is the throughput limiter.
// ---------------------------------------------------------------------------
__global__ __launch_bounds__(BLOCK_THREADS) void rank1_attn_kernel(
    const float* __restrict__ x,
    const float* __restrict__ w1p, const float* __restrict__ b1p,
    const float* __restrict__ w2p, const float* __restrict__ b2p,
    const float* __restrict__ bounds,
    float* __restrict__ out) {
    __shared__ float xs[NN];

    const int tid = threadIdx.x;
    const int waveId = tid >> 5;
    const int lane = tid & 31;
    const int tile = blockIdx.x * WAVES_PER_BLOCK + waveId;
    const int b = tile / TILES_PER_BATCH;          // 25 blocks per batch: block never straddles batches
    const int m0 = (tile % TILES_PER_BATCH) * 16;

    const float w1 = *w1p, b1 = *b1p, w2 = *w2p, b2 = *b2p;

    // Stage this batch's x row (12.8 KB) in LDS once per block.
    const float* xb = x + b * NN;
    for (int i = tid; i < NN; i += BLOCK_THREADS) xs[i] = xb[i];
    __syncthreads();

    // Analytic row-max from per-batch bounds of c = w2*x + b2.
    const float xmn = bounds[2 * b + 0];
    const float xmx = bounds[2 * b + 1];
    const float cA = w2 * xmx + b2;
    const float cB = w2 * xmn + b2;
    const float cmax = fmaxf(cA, cB);
    const float cmin = fminf(cA, cB);

    const int half = lane >> 4;   // K half for A/B operands
    const int j = lane & 15;      // A row within tile; B/D column
    const int m = m0 + j;

    const float xm = xs[m];
    const float t = w1 * xm + b1;                       // a_m
    const float rowmax = (t >= 0.0f) ? t * cmax : t * cmin;

    const bool selx = (j == 0);
    const bool sel1 = (j == 1);

    v8f acc = {};
#pragma unroll 4
    for (int nb = 0; nb < NN; nb += 4) {
        // lanes 0-15 handle K=0,1 (n = nb, nb+1); lanes 16-31 handle K=2,3
        const float2 xv = *(const float2*)(&xs[nb + 2 * half]);
        const float c0 = w2 * xv.x + b2;
        const float c1 = w2 * xv.y + b2;
        v2f A;
        A.x = __expf(t * c0 - rowmax);
        A.y = __expf(t * c1 - rowmax);
        v2f Bv;
        Bv.x = selx ? xv.x : (sel1 ? 1.0f : 0.0f);
        Bv.y = selx ? xv.y : (sel1 ? 1.0f : 0.0f);
        acc = __builtin_amdgcn_wmma_f32_16x16x4_f32(
            /*neg_a=*/false, A, /*neg_b=*/false, Bv,
            /*c_mod=*/(short)0, acc, /*reuse_a=*/false, /*reuse_b=*/false);
    }

    // D layout: row r (0-7) in acc[r] lanes 0-15 (col=lane), rows 8-15 lanes 16-31.
    // num = column 0 (lanes 0 / 16), den = column 1 (lanes 1 / 17).
    float num = 0.0f, den = 1.0f;
#pragma unroll
    for (int r = 0; r < 8; ++r) {
        const float nlo = __shfl(acc[r], 0, 32);
        const float nhi = __shfl(acc[r], 16, 32);
        const float dlo = __shfl(acc[r], 1, 32);
        const float dhi = __shfl(acc[r], 17, 32);
        if ((j & 7) == r) {
            const bool hi = (j >= 8);
            num = hi ? nhi : nlo;
            den = hi ? dhi : dlo;
        }
    }

    if (half == 0) {
        out[b * NN + m] = num / den + xm;   // residual add
    }
}

extern "C" void kernel_launch(void* const* d_in, const int* in_sizes, int n_in,
                              void* d_out, int out_size, void* d_ws, size_t ws_size,
                              hipStream_t stream) {
    const float* x  = (const float*)d_in[0];
    const float* w1 = (const float*)d_in[1];
    const float* b1 = (const float*)d_in[2];
    const float* w2 = (const float*)d_in[3];
    const float* b2 = (const float*)d_in[4];
    float* out = (float*)d_out;
    float* bounds = (float*)d_ws;   // 16 * 2 floats

    rowstat_kernel<<<NBATCH, 32, 0, stream>>>(x, bounds);

    const int total_tiles = NBATCH * TILES_PER_BATCH;            // 3200
    const int blocks = total_tiles / WAVES_PER_BLOCK;            // 400
    rank1_attn_kernel<<<blocks, BLOCK_THREADS, 0, stream>>>(
        x, w1, b1, w2, b2, bounds, out);
}